// GraphletEncoder_90941637525518
// MI455X (gfx1250) — compile-verified
//
#include <hip/hip_runtime.h>
#include <hip/hip_bf16.h>

typedef float v2f __attribute__((ext_vector_type(2)));
typedef float v8f __attribute__((ext_vector_type(8)));

#define DIM 64

// ---------------------------------------------------------------------------
// Utility: fill a float buffer with a constant
// ---------------------------------------------------------------------------
__global__ void fill_f32(float* __restrict__ p, float v, int n) {
    int i = blockIdx.x * blockDim.x + threadIdx.x;
    if (i < n) p[i] = v;
}

// ---------------------------------------------------------------------------
// Degree accumulation: deg[dst[e]] += 1  (deg pre-initialized to 1.0 = self loop)
// ---------------------------------------------------------------------------
__global__ void deg_scatter(const int* __restrict__ dst, float* __restrict__ deg, int E) {
    int e = blockIdx.x * blockDim.x + threadIdx.x;
    if (e < E) atomicAdd(&deg[dst[e]], 1.0f);
}

// dinv[i] = rsqrt(deg[i])  (in place; deg >= 1 always)
__global__ void dinv_kernel(float* __restrict__ deg, int n) {
    int i = blockIdx.x * blockDim.x + threadIdx.x;
    if (i < n) deg[i] = rsqrtf(deg[i]);
}

// ---------------------------------------------------------------------------
// WMMA GEMM: H[M x 64] = X[M x 64] @ W[64 x 64], fp32 via V_WMMA_F32_16X16X4_F32
// Block = 128 threads (4 waves). Each wave owns one 16x16 output tile:
//   blockIdx.x -> row tile (16 rows), wave id -> column tile (4 tiles = 64 cols).
// M must be a multiple of 16 (N_NODES = 100000 = 6250*16). EXEC all-ones.
//
// Fragment layouts per CDNA5 ISA 7.12.2 (32-bit, wave32):
//   A (16x4):  lanes 0-15 rows M=0..15 with K={k,k+1}, lanes 16-31 K={k+2,k+3}
//   B (4x16):  lanes hold N columns; K split across lane halves (mirror of A)
//   C/D:       VGPR v: lanes 0-15 -> (M=v, N=lane), lanes 16-31 -> (M=v+8, N=lane-16)
// ---------------------------------------------------------------------------
__global__ void gemm64_wmma(const float* __restrict__ X, const float* __restrict__ W,
                            float* __restrict__ H, int M) {
    const int wave = threadIdx.x >> 5;        // 0..3 -> column tile
    const int lane = threadIdx.x & 31;
    const int half = lane >> 4;               // 0: K lo pair, 1: K hi pair
    const int l    = lane & 15;
    const int row0 = blockIdx.x * 16;
    const int col0 = wave * 16;
    if (row0 >= M) return;

    v8f c = {};
    const float* __restrict__ xrow = X + (size_t)(row0 + l) * DIM;

    #pragma unroll
    for (int k = 0; k < DIM; k += 4) {
        v2f a, b;
        a.x = xrow[k + 2 * half + 0];
        a.y = xrow[k + 2 * half + 1];
        b.x = W[(size_t)(k + 2 * half + 0) * DIM + col0 + l];
        b.y = W[(size_t)(k + 2 * half + 1) * DIM + col0 + l];
        c = __builtin_amdgcn_wmma_f32_16x16x4_f32(
                /*neg_a=*/false, a, /*neg_b=*/false, b,
                /*c_mod=*/(short)0, c, /*reuse_a=*/false, /*reuse_b=*/false);
    }

    #pragma unroll
    for (int v = 0; v < 8; ++v) {
        int row = row0 + v + 8 * half;
        H[(size_t)row * DIM + col0 + l] = c[v];
    }
}

// ---------------------------------------------------------------------------
// Aggregation init with self-loop term: agg[i][f] = dinv[i]^2 * h[i][f]
// Vectorized float4 (row stride 256B, f0 multiple of 16B -> aligned).
// ---------------------------------------------------------------------------
__global__ void agg_init(const float* __restrict__ h, const float* __restrict__ dinv,
                         float* __restrict__ agg, int n) {
    int t = blockIdx.x * blockDim.x + threadIdx.x;   // n * 16 threads
    if (t >= n * 16) return;
    int i  = t >> 4;
    int f0 = (t & 15) * 4;
    float d  = dinv[i];
    float nn = d * d;
    const float4 v = *(const float4*)(h + (size_t)i * DIM + f0);
    float4 o;
    o.x = nn * v.x; o.y = nn * v.y; o.z = nn * v.z; o.w = nn * v.w;
    *(float4*)(agg + (size_t)i * DIM + f0) = o;
}

// ---------------------------------------------------------------------------
// Edge scatter: agg[dst][f] += dinv[src]*dinv[dst] * h[src][f]
// 16 lanes per edge, float4 gather (L2-resident), 4 f32 atomics per lane.
// ---------------------------------------------------------------------------
__global__ void edge_scatter(const int* __restrict__ src, const int* __restrict__ dst,
                             const float* __restrict__ h, const float* __restrict__ dinv,
                             float* __restrict__ agg, int E) {
    long long t = (long long)blockIdx.x * blockDim.x + threadIdx.x;  // E * 16 threads
    if (t >= (long long)E * 16) return;
    int e  = (int)(t >> 4);
    int f0 = ((int)t & 15) * 4;
    int s = src[e], d = dst[e];
    float nrm = dinv[s] * dinv[d];
    const float4 v = *(const float4*)(h + (size_t)s * DIM + f0);
    float* base = agg + (size_t)d * DIM + f0;
    atomicAdd(base + 0, nrm * v.x);
    atomicAdd(base + 1, nrm * v.y);
    atomicAdd(base + 2, nrm * v.z);
    atomicAdd(base + 3, nrm * v.w);
}

// ---------------------------------------------------------------------------
// Bias + ReLU: out[i][f] = max(agg[i][f] + b[f], 0)
// ---------------------------------------------------------------------------
__global__ void bias_relu(const float* __restrict__ agg, const float* __restrict__ b,
                          float* __restrict__ out, int n) {
    int t = blockIdx.x * blockDim.x + threadIdx.x;
    if (t >= n * DIM) return;
    int f = t & (DIM - 1);
    out[t] = fmaxf(agg[t] + b[f], 0.0f);
}

// ---------------------------------------------------------------------------
// Pooling: counts[g] += 1 per node; pooled[g][f] += h[i][f]
// ---------------------------------------------------------------------------
__global__ void pool_count(const int* __restrict__ batch, float* __restrict__ counts, int n) {
    int i = blockIdx.x * blockDim.x + threadIdx.x;
    if (i < n) atomicAdd(&counts[batch[i]], 1.0f);
}

__global__ void pool_sum(const int* __restrict__ batch, const float* __restrict__ h,
                         float* __restrict__ pooled, int n) {
    int t = blockIdx.x * blockDim.x + threadIdx.x;   // n * DIM threads
    if (t >= n * DIM) return;
    int i = t >> 6;
    int f = t & (DIM - 1);
    atomicAdd(&pooled[(size_t)batch[i] * DIM + f], h[t]);
}

__global__ void pool_finalize(const float* __restrict__ pooled, const float* __restrict__ counts,
                              float* __restrict__ out, int G) {
    int t = blockIdx.x * blockDim.x + threadIdx.x;   // G * DIM threads
    if (t >= G * DIM) return;
    int g = t >> 6;
    out[t] = pooled[t] / fmaxf(counts[g], 1.0f);
}

// ---------------------------------------------------------------------------
// Launch
// ---------------------------------------------------------------------------
extern "C" void kernel_launch(void* const* d_in, const int* in_sizes, int n_in,
                              void* d_out, int out_size, void* d_ws, size_t ws_size,
                              hipStream_t stream) {
    const float* x      = (const float*)d_in[0];
    const float* W1     = (const float*)d_in[1];
    const float* b1     = (const float*)d_in[2];
    const float* W2     = (const float*)d_in[3];
    const float* b2     = (const float*)d_in[4];
    const int*   edge   = (const int*)d_in[5];     // [2, E] flat
    const int*   batch  = (const int*)d_in[6];     // [N]
    float*       out    = (float*)d_out;

    const int N = in_sizes[0] / DIM;       // 100000
    const int E = in_sizes[5] / 2;         // 1600000
    const int G = out_size / DIM;          // 1000

    const int* srcArr = edge;
    const int* dstArr = edge + E;

    // Workspace layout (floats)
    float* ws     = (float*)d_ws;
    float* dinv   = ws;                    // N       (deg -> dinv in place)
    float* bufA   = dinv + N;              // N*DIM
    float* bufB   = bufA + (size_t)N * DIM;// N*DIM
    float* pooled = bufB + (size_t)N * DIM;// G*DIM
    float* counts = pooled + (size_t)G * DIM; // G
    (void)ws_size; (void)n_in;

    const int T = 256;
    auto blocks = [&](long long n) { return (unsigned)((n + T - 1) / T); };

    // --- degrees / normalization ---
    fill_f32<<<blocks(N), T, 0, stream>>>(dinv, 1.0f, N);          // self loops
    fill_f32<<<blocks(G * DIM), T, 0, stream>>>(pooled, 0.0f, G * DIM);
    fill_f32<<<blocks(G), T, 0, stream>>>(counts, 0.0f, G);
    deg_scatter<<<blocks(E), T, 0, stream>>>(dstArr, dinv, E);
    dinv_kernel<<<blocks(N), T, 0, stream>>>(dinv, N);

    const unsigned gemmBlocks = (unsigned)(N / 16);                // N multiple of 16

    // --- layer 1: A = x @ W1 ; B = Ahat*A ; A = relu(B + b1) ---
    gemm64_wmma<<<gemmBlocks, 128, 0, stream>>>(x, W1, bufA, N);
    agg_init<<<blocks((long long)N * 16), T, 0, stream>>>(bufA, dinv, bufB, N);
    edge_scatter<<<blocks((long long)E * 16), T, 0, stream>>>(srcArr, dstArr, bufA, dinv, bufB, E);
    bias_relu<<<blocks((long long)N * DIM), T, 0, stream>>>(bufB, b1, bufA, N);

    // --- layer 2: B = A @ W2 ; A = Ahat*B ; B = relu(A + b2) ---
    gemm64_wmma<<<gemmBlocks, 128, 0, stream>>>(bufA, W2, bufB, N);
    agg_init<<<blocks((long long)N * 16), T, 0, stream>>>(bufB, dinv, bufA, N);
    edge_scatter<<<blocks((long long)E * 16), T, 0, stream>>>(srcArr, dstArr, bufB, dinv, bufA, E);
    bias_relu<<<blocks((long long)N * DIM), T, 0, stream>>>(bufA, b2, bufB, N);

    // --- mean pool per graph ---
    pool_count<<<blocks(N), T, 0, stream>>>(batch, counts, N);
    pool_sum<<<blocks((long long)N * DIM), T, 0, stream>>>(batch, bufB, pooled, N);
    pool_finalize<<<blocks((long long)G * DIM), T, 0, stream>>>(pooled, counts, out, G);
}